// binary_constrastive_86225763434864
// MI455X (gfx1250) — compile-verified
//
#include <hip/hip_runtime.h>

typedef __attribute__((ext_vector_type(2))) float v2f;
typedef __attribute__((ext_vector_type(8))) float v8f;

#define TEMP_INV 10.0f     // 1/temp, temp = 0.1
#define EPS_F    1e-5f

// ---------------------------------------------------------------------------
// Kernel 1: WMMA-based batched dot products.
// One wave handles one (16-row tile, K-half) pair.
//   A  = 16x4 fp32 chunk of rows r0..r0+15 of x
//   B0 = 4x16 chunk of the same rows        -> diag(acc0) = <x_i, x_i>
//   B1 = 4x16 chunk of gathered rows        -> diag(acc1) = <x_i, x_s(i)>
// Lane (m = lane&15, kh = lane>>4) holds a contiguous float2 at column
// k + 2*kh of its row for both A and B operands (per ISA 7.12.2 layouts).
// ---------------------------------------------------------------------------
__global__ __launch_bounds__(256) void bc_dots(const float* __restrict__ x,
                                               const int* __restrict__ samples,
                                               float* __restrict__ pdots, // [4*B]
                                               int B, int D)
{
    const int wave  = threadIdx.x >> 5;
    const int lane  = threadIdx.x & 31;
    const int wgid  = blockIdx.x * 8 + wave;   // global wave id
    const int tile  = wgid >> 1;               // 16-row tile
    const int kpart = wgid & 1;                // which K half
    const int r0    = tile * 16;
    const int m     = lane & 15;
    const int kh    = lane >> 4;

    const int Dh = D >> 1;
    const int k0 = kpart * Dh;

    const int srow = samples[r0 + m];
    const float* xr  = x + (size_t)(r0 + m) * (size_t)D + (size_t)k0 + 2 * kh;
    const float* xsr = x + (size_t)srow     * (size_t)D + (size_t)k0 + 2 * kh;

    v8f acc0 = {};  // X  * X^T  tile (need diagonal)
    v8f acc1 = {};  // X  * Xs^T tile (need diagonal)

    for (int k = 0; k < Dh; k += 32) {
#pragma unroll
        for (int u = 0; u < 8; ++u) {
            v2f a  = *(const v2f*)(xr  + k + 4 * u);   // A chunk (also B for sim0)
            v2f bs = *(const v2f*)(xsr + k + 4 * u);   // B chunk for sim1
            acc0 = __builtin_amdgcn_wmma_f32_16x16x4_f32(
                       false, a, false, a,  (short)0, acc0, false, false);
            acc1 = __builtin_amdgcn_wmma_f32_16x16x4_f32(
                       false, a, false, bs, (short)0, acc1, false, false);
        }
    }

    // C/D layout: VGPR v, lanes 0-15 -> (M=v, N=lane); lanes 16-31 -> (M=v+8, N=lane-16).
    // Diagonal (i,i): i in [0,8)  -> lane i,   acc[i]
    //                 i in [8,16) -> lane i+16, acc[i-8]
    const int sel = (lane < 8) ? lane : (lane - 24);  // negative for lanes 8..23 -> no match
    float s0 = 0.f, s1 = 0.f;
#pragma unroll
    for (int j = 0; j < 8; ++j) {
        s0 = (sel == j) ? acc0[j] : s0;
        s1 = (sel == j) ? acc1[j] : s1;
    }

    const bool active = (lane < 8) || (lane >= 24);
    const int  irow   = r0 + ((lane < 8) ? lane : (lane - 16));
    if (active) {
        pdots[(size_t)(0 + kpart) * (size_t)B + irow] = s0;  // sim0 halves in [0, 2B)
        pdots[(size_t)(2 + kpart) * (size_t)B + irow] = s1;  // sim1 halves in [2B, 4B)
    }
}

// ---------------------------------------------------------------------------
// Kernel 2: per-row BCE loss + fixed-order block reduction.
// ---------------------------------------------------------------------------
__global__ __launch_bounds__(256) void bc_loss(const float* __restrict__ pdots,
                                               const int* __restrict__ y,
                                               const int* __restrict__ samples,
                                               float* __restrict__ bsum, int B)
{
    __shared__ float sm[256];
    const int i = blockIdx.x * 256 + threadIdx.x;

    float s0 = pdots[i] + pdots[(size_t)B + i];
    float s1 = pdots[(size_t)2 * B + i] + pdots[(size_t)3 * B + i];
    float diff = (s0 - s1) * TEMP_INV;                 // (d0 - d1)/temp
    float lab  = (y[i] != y[samples[i]]) ? 1.0f : 0.0f;
    float p    = 1.0f / (1.0f + __expf(-diff));
    float loss = -__logf(p + EPS_F) * lab - __logf(1.0f - p + EPS_F) * (1.0f - lab);

    sm[threadIdx.x] = loss;
    __syncthreads();
    for (int off = 128; off > 0; off >>= 1) {
        if ((int)threadIdx.x < off) sm[threadIdx.x] += sm[threadIdx.x + off];
        __syncthreads();
    }
    if (threadIdx.x == 0) bsum[blockIdx.x] = sm[0];
}

// ---------------------------------------------------------------------------
// Kernel 3: final mean over block sums.
// ---------------------------------------------------------------------------
__global__ __launch_bounds__(256) void bc_final(const float* __restrict__ bsum,
                                                float* __restrict__ out,
                                                int nb, float invB)
{
    __shared__ float sm[256];
    float s = 0.f;
    for (int t = threadIdx.x; t < nb; t += 256) s += bsum[t];
    sm[threadIdx.x] = s;
    __syncthreads();
    for (int off = 128; off > 0; off >>= 1) {
        if ((int)threadIdx.x < off) sm[threadIdx.x] += sm[threadIdx.x + off];
        __syncthreads();
    }
    if (threadIdx.x == 0) out[0] = sm[0] * invB;
}

// ---------------------------------------------------------------------------
extern "C" void kernel_launch(void* const* d_in, const int* in_sizes, int n_in,
                              void* d_out, int out_size, void* d_ws, size_t ws_size,
                              hipStream_t stream)
{
    const float* x       = (const float*)d_in[0];
    const int*   y       = (const int*)d_in[1];
    const int*   samples = (const int*)d_in[2];

    const int B = in_sizes[1];            // 16384
    const int D = in_sizes[0] / B;        // 2048

    float* pdots = (float*)d_ws;                       // 4*B floats
    float* bsum  = pdots + (size_t)4 * (size_t)B;      // B/256 floats

    const int waves   = (B / 16) * 2;     // 2 K-halves per 16-row tile
    const int blocks1 = waves / 8;        // 8 waves per 256-thread block
    const int blocks2 = B / 256;

    bc_dots <<<blocks1, 256, 0, stream>>>(x, samples, pdots, B, D);
    bc_loss <<<blocks2, 256, 0, stream>>>(pdots, y, samples, bsum, B);
    bc_final<<<1,       256, 0, stream>>>(bsum, (float*)d_out, blocks2, 1.0f / (float)B);
}